// AdaptiveVoxelization_27023934226795
// MI455X (gfx1250) — compile-verified
//
#include <hip/hip_runtime.h>
#include <hip/hip_bf16.h>
#include <stdint.h>

// ---------------------------------------------------------------------------
// AdaptiveVoxelization for MI455X (gfx1250)
// Memory-bound streaming workload: ~128 MB total traffic -> ~5.5us @ 23.3TB/s.
// CDNA5 async global->LDS DMA (ASYNCcnt) streams the point cloud through a
// double-buffered LDS tile; all outputs use non-temporal stores; the zyx
// int32 triple per point is packed into a single b96 store.
// ---------------------------------------------------------------------------

#define RES   100
#define RES3  (RES * RES * RES)

// 3 x int32, natural size 12B, forced 4B alignment so a single unaligned-ok
// global_store_b96 is emitted at out + 3*i.
typedef int int3v __attribute__((ext_vector_type(3), aligned(4)));

// ---- CDNA5 async global->LDS staging (one b128 per lane per call) ---------
__device__ __forceinline__ void async_stage_b128(const void* gsrc, void* lds_dst) {
    // Low 32 bits of a flat shared-memory address are the wave-relative LDS
    // byte offset (aperture lives in the high bits).
    unsigned ldsoff = (unsigned)(unsigned long long)lds_dst;
    asm volatile("global_load_async_to_lds_b128 %0, %1, off th:TH_LOAD_NT"
                 :: "v"(ldsoff), "v"(gsrc)
                 : "memory");
}

__device__ __forceinline__ void wait_async_le1() {
    asm volatile("s_wait_asynccnt 0x1" ::: "memory");
}
__device__ __forceinline__ void wait_async_le0() {
    asm volatile("s_wait_asynccnt 0x0" ::: "memory");
}

// ---- per-point voxelization (mmdet3d dynamic_voxelize semantics) ----------
__device__ __forceinline__ void voxelize_one(float px, float py, float pz,
                                             int* __restrict__ out, long i) {
    // voxel_size = (0.1, 0.1, 0.2), cmin = (-50, -50, -1)
    // grid = round((max-min)/vs) = (2000, 1000, 20)
    float cx = floorf((px - (-50.0f)) / 0.1f);
    float cy = floorf((py - (-50.0f)) / 0.1f);
    float cz = floorf((pz - (-1.0f)) / 0.2f);
    bool valid = (cx >= 0.0f) && (cx < 2000.0f) &&
                 (cy >= 0.0f) && (cy < 1000.0f) &&
                 (cz >= 0.0f) && (cz < 20.0f);
    int3v v;
    v.x = valid ? (int)cz : -1;   // zyx order
    v.y = valid ? (int)cy : -1;
    v.z = valid ? (int)cx : -1;
    // single-pass output -> one non-temporal b96 store per point
    __builtin_nontemporal_store(v, (int3v*)(out + 3 * i));
}

// ---- Kernel 1: stream points via double-buffered async LDS tiles ----------
__global__ void __launch_bounds__(256)
voxelize_kernel(const float4* __restrict__ p4, int* __restrict__ out, int npts) {
    __shared__ float4 buf[2][256];
    const int tid     = threadIdx.x;
    const int ntiles  = npts >> 8;        // 256 points per tile
    const int tstride = gridDim.x;
    int t = blockIdx.x;

    // Preload first tile for this block.
    if (t < ntiles) {
        async_stage_b128(p4 + ((long)t << 8) + tid, &buf[0][tid]);
    }

    int b = 0;
    while (t < ntiles) {
        const int tn = t + tstride;
        if (tn < ntiles) {
            // Kick off next tile, then wait until only it is outstanding.
            async_stage_b128(p4 + ((long)tn << 8) + tid, &buf[b ^ 1][tid]);
            wait_async_le1();            // async loads complete in order
        } else {
            wait_async_le0();
        }
        // Each lane consumes exactly the slot it DMA'd -> no barrier needed.
        float4 p = buf[b][tid];
        voxelize_one(p.x, p.y, p.z, out, ((long)t << 8) + tid);
        b ^= 1;
        t = tn;
    }

    // Generic tail (npts not a multiple of 256); empty for N = 4,000,000.
    const int done = ntiles << 8;
    const int gsz  = gridDim.x * blockDim.x;
    for (int i = done + blockIdx.x * blockDim.x + tid; i < npts; i += gsz) {
        float4 p = p4[i];
        voxelize_one(p.x, p.y, p.z, out, i);
    }
}

// ---- Kernel 2: adaptive voxel centers (index-generated, store-only) -------
__global__ void __launch_bounds__(256)
centers_kernel(float* __restrict__ cen, float* __restrict__ msk) {
    int n = blockIdx.x * blockDim.x + threadIdx.x;
    if (n >= RES3) return;

    int k = n % RES;            // z index (fastest, 'ij' meshgrid ravel)
    int j = (n / RES) % RES;    // y index
    int i = n / (RES * RES);    // x index

    // linspace over point_cloud_range per dim
    const float sx = 200.0f / 99.0f;   // (150 - -50)/(R-1)
    const float sy = 100.0f / 99.0f;   // ( 50 - -50)/(R-1)
    const float sz =   4.0f / 99.0f;   // (  3 -  -1)/(R-1)
    float x = -50.0f + (float)i * sx;
    float y = -50.0f + (float)j * sy;
    float z =  -1.0f + (float)k * sz;

    // quadratic mode: sf = |n|^2 / norm_c^2 ; grid is a full Cartesian
    // product so norm_c = sqrt(150^2 + 50^2 + 3^2) = sqrt(25009) (corner).
    float nrm = sqrtf(x * x + y * y + z * z);
    const float nc = sqrtf(25009.0f);
    float sf = (nrm * nrm) / (nc * nc);
    float scale = 1.0f + sf * 25.0f;   // SCALING = 25

    float cx = x * scale, cy = y * scale, cz = z * scale;
    bool m = (cx >= -50.0f) && (cx <= 150.0f) &&
             (cy >= -50.0f) && (cy <=  50.0f) &&
             (cz >=  -1.0f) && (cz <=   3.0f);

    __builtin_nontemporal_store(cx, cen + 0 * RES3 + n);
    __builtin_nontemporal_store(cy, cen + 1 * RES3 + n);
    __builtin_nontemporal_store(cz, cen + 2 * RES3 + n);
    __builtin_nontemporal_store(m ? 1.0f : 0.0f, msk + n);
}

// ---------------------------------------------------------------------------
// d_out layout (concatenated flat, return order, 4-byte elements):
//   [0, 3N)                 coors   int32 (zyx, -1 invalid)
//   [3N, 3N+3*R^3)          centers float32, [3, R^3]
//   [3N+3*R^3, 3N+4*R^3)    mask    as 1.0f / 0.0f
// ---------------------------------------------------------------------------
extern "C" void kernel_launch(void* const* d_in, const int* in_sizes, int n_in,
                              void* d_out, int out_size, void* d_ws, size_t ws_size,
                              hipStream_t stream) {
    (void)n_in; (void)out_size; (void)d_ws; (void)ws_size;

    const float* pts = (const float*)d_in[0];
    const int npts = in_sizes[0] / 4;            // [N,4] -> N

    int*   coors   = (int*)d_out;
    float* centers = (float*)d_out + (size_t)npts * 3;
    float* maskout = centers + (size_t)3 * RES3;

    const int ntiles = npts >> 8;
    int blocks = 2048;                           // persistent blocks: 16K waves
    if (ntiles > 0 && ntiles < blocks) blocks = ntiles;
    if (blocks == 0) blocks = 1;

    voxelize_kernel<<<blocks, 256, 0, stream>>>(
        (const float4*)pts, coors, npts);

    centers_kernel<<<(RES3 + 255) / 256, 256, 0, stream>>>(centers, maskout);
}